// Interaction_GraphConvolution_9131100472115
// MI455X (gfx1250) — compile-verified
//
#include <hip/hip_runtime.h>
#include <hip/hip_bf16.h>

typedef float v2f __attribute__((ext_vector_type(2)));
typedef float v8f __attribute__((ext_vector_type(8)));

#define GN   2048
#define FIN  128
#define FOUT 64
#define JT2  32                 // j rows per workgroup = 2 WMMA row-tiles
#define KT   64                 // k columns per pass    = 4 WMMA col-tiles
// LDS A-slab row stride (floats): 2052 % 64 == 4 -> lane m at bank base 4m,
// +2-dword offset for lanes 16-31 -> conflict-free dual-dword A loads.
// 2052*4 = 8208 = 16*513 -> rows stay 16B-aligned for float4 staging.
#define ASTRIDE (GN + 4)
#define LDS_BYTES (JT2 * ASTRIDE * 4)   // 262,656 B (<= 320KB WGP limit)

// Phase 1: Ht[f, i] = (X @ W)[i, f] + bias[f]   (transposed, [64, 2048])
__global__ __launch_bounds__(256)
void h_gemm_kernel(const float* __restrict__ X, const float* __restrict__ W,
                   const float* __restrict__ bias, float* __restrict__ Ht) {
  int idx = blockIdx.x * 256 + threadIdx.x;   // 64*2048 threads
  int f = idx >> 11;
  int i = idx & (GN - 1);
  if (f >= FOUT) return;
  float s = bias[f];
#pragma unroll 8
  for (int c = 0; c < FIN; ++c)
    s = fmaf(X[(size_t)i * FIN + c], W[c * FOUT + f], s);
  Ht[(size_t)f * GN + i] = s;
}

// Phase 2: out[j,f] = sum_k [ (A diag(h_f)) @ mh ][j,k] * mf[j,k] * h_f[k]
// Workgroup = 8 waves = one 32-row j-slab; wave w owns feature f = 8*by + w
// and computes a 2x4 grid of 16x16 WMMA tiles. h_f is folded into the
// A-fragment (reused across all 4 k-tiles), so B-fragments are raw mh loads.
__global__ __launch_bounds__(256)
void gconv_kernel(const float* __restrict__ A, const float* __restrict__ mh,
                  const float* __restrict__ mf, const float* __restrict__ Ht,
                  float* __restrict__ out) {
  extern __shared__ float sA[];              // [JT2][ASTRIDE]

  const int tid  = threadIdx.x;
  const int wave = tid >> 5;
  const int lane = tid & 31;
  const int j0   = blockIdx.x * JT2;
  const int f    = blockIdx.y * 8 + wave;

  // Stage A[j0:j0+32, :] into LDS, coalesced float4 (64 iters * 256 thr).
  for (int e = tid; e < JT2 * (GN / 4); e += 256) {
    int row = e >> 9;                        // / 512
    int c4  = e & 511;
    float4 v = ((const float4*)(A + (size_t)(j0 + row) * GN))[c4];
    *(float4*)&sA[(size_t)row * ASTRIDE + c4 * 4] = v;
  }
  __syncthreads();

  // WMMA f32 16x16x4 operand mapping (wave32):
  //  A-frag lane: M = lane&15, VGPR0/1 = K = 2*(lane>>4) + {0,1}
  //  B-frag lane: N = lane&15, VGPR0/1 = K = 2*(lane>>4) + {0,1}
  //  D lane:      rows M = v + 8*(lane>>4) (v=0..7), col N = lane&15
  const int l16  = lane & 15;
  const int hi   = lane >> 4;
  const int koff = hi * 2;
  const float* __restrict__ hf   = Ht + (size_t)f * GN;
  const float* __restrict__ sAr0 = &sA[(size_t)l16 * ASTRIDE];
  const float* __restrict__ sAr1 = &sA[(size_t)(16 + l16) * ASTRIDE];

  float S0[8] = {0.f,0.f,0.f,0.f,0.f,0.f,0.f,0.f};
  float S1[8] = {0.f,0.f,0.f,0.f,0.f,0.f,0.f,0.f};

  for (int k0 = 0; k0 < GN; k0 += KT) {
    v8f acc00 = {}, acc01 = {}, acc02 = {}, acc03 = {};
    v8f acc10 = {}, acc11 = {}, acc12 = {}, acc13 = {};
    const float* __restrict__ mhp = mh + (size_t)koff * GN + (k0 + l16);
    const float* __restrict__ hp  = hf + koff;
#pragma unroll 2
    for (int i = 0; i < GN; i += 4) {
      float h0 = hp[i];
      float h1 = hp[i + 1];
      v2f a0, a1;                             // A' = A * h_f (per-K scaling)
      a0.x = h0 * sAr0[i + koff];  a0.y = h1 * sAr0[i + koff + 1];
      a1.x = h0 * sAr1[i + koff];  a1.y = h1 * sAr1[i + koff + 1];
      v2f b0, b1, b2, b3;                     // raw mh, 4 col-tiles, 2 rows
      b0.x = mhp[(size_t)i * GN];       b0.y = mhp[(size_t)i * GN + GN];
      b1.x = mhp[(size_t)i * GN + 16];  b1.y = mhp[(size_t)i * GN + GN + 16];
      b2.x = mhp[(size_t)i * GN + 32];  b2.y = mhp[(size_t)i * GN + GN + 32];
      b3.x = mhp[(size_t)i * GN + 48];  b3.y = mhp[(size_t)i * GN + GN + 48];
      acc00 = __builtin_amdgcn_wmma_f32_16x16x4_f32(false, a0, false, b0, (short)0, acc00, false, false);
      acc01 = __builtin_amdgcn_wmma_f32_16x16x4_f32(false, a0, false, b1, (short)0, acc01, false, false);
      acc02 = __builtin_amdgcn_wmma_f32_16x16x4_f32(false, a0, false, b2, (short)0, acc02, false, false);
      acc03 = __builtin_amdgcn_wmma_f32_16x16x4_f32(false, a0, false, b3, (short)0, acc03, false, false);
      acc10 = __builtin_amdgcn_wmma_f32_16x16x4_f32(false, a1, false, b0, (short)0, acc10, false, false);
      acc11 = __builtin_amdgcn_wmma_f32_16x16x4_f32(false, a1, false, b1, (short)0, acc11, false, false);
      acc12 = __builtin_amdgcn_wmma_f32_16x16x4_f32(false, a1, false, b2, (short)0, acc12, false, false);
      acc13 = __builtin_amdgcn_wmma_f32_16x16x4_f32(false, a1, false, b3, (short)0, acc13, false, false);
    }
    // Epilogue: weight each 16x16 tile by mf[j,k]*h_f[k] and fold over k.
    float hk0 = hf[k0 + l16];
    float hk1 = hf[k0 + 16 + l16];
    float hk2 = hf[k0 + 32 + l16];
    float hk3 = hf[k0 + 48 + l16];
    const float* __restrict__ mfp0 = mf + (size_t)(j0 + hi * 8) * GN + (k0 + l16);
    const float* __restrict__ mfp1 = mf + (size_t)(j0 + 16 + hi * 8) * GN + (k0 + l16);
#pragma unroll
    for (int v = 0; v < 8; ++v) {
      float s0 = S0[v], s1 = S1[v];
      s0 = fmaf(acc00[v] * hk0, mfp0[(size_t)v * GN],      s0);
      s0 = fmaf(acc01[v] * hk1, mfp0[(size_t)v * GN + 16], s0);
      s0 = fmaf(acc02[v] * hk2, mfp0[(size_t)v * GN + 32], s0);
      s0 = fmaf(acc03[v] * hk3, mfp0[(size_t)v * GN + 48], s0);
      s1 = fmaf(acc10[v] * hk0, mfp1[(size_t)v * GN],      s1);
      s1 = fmaf(acc11[v] * hk1, mfp1[(size_t)v * GN + 16], s1);
      s1 = fmaf(acc12[v] * hk2, mfp1[(size_t)v * GN + 32], s1);
      s1 = fmaf(acc13[v] * hk3, mfp1[(size_t)v * GN + 48], s1);
      S0[v] = s0; S1[v] = s1;
    }
  }

  // Reduce the 16 column-lanes within each half-wave.
#pragma unroll
  for (int v = 0; v < 8; ++v) {
    float s0 = S0[v], s1 = S1[v];
    s0 += __shfl_xor(s0, 1, 32);  s1 += __shfl_xor(s1, 1, 32);
    s0 += __shfl_xor(s0, 2, 32);  s1 += __shfl_xor(s1, 2, 32);
    s0 += __shfl_xor(s0, 4, 32);  s1 += __shfl_xor(s1, 4, 32);
    s0 += __shfl_xor(s0, 8, 32);  s1 += __shfl_xor(s1, 8, 32);
    S0[v] = s0; S1[v] = s1;
  }
  if (l16 == 0) {
#pragma unroll
    for (int v = 0; v < 8; ++v) {
      out[(size_t)(j0 + hi * 8 + v) * FOUT + f]      = S0[v];
      out[(size_t)(j0 + 16 + hi * 8 + v) * FOUT + f] = S1[v];
    }
  }
}

extern "C" void kernel_launch(void* const* d_in, const int* in_sizes, int n_in,
                              void* d_out, int out_size, void* d_ws, size_t ws_size,
                              hipStream_t stream) {
  (void)in_sizes; (void)n_in; (void)out_size; (void)ws_size;
  const float* X  = (const float*)d_in[0];   // node_features   [2048,128]
  const float* A  = (const float*)d_in[1];   // adjacency       [2048,2048]
  const float* mf = (const float*)d_in[2];   // mask_father     [2048,2048]
  const float* mh = (const float*)d_in[3];   // mask_hadamard   [2048,2048]
  const float* W  = (const float*)d_in[4];   // weight          [128,64]
  const float* b  = (const float*)d_in[5];   // bias            [64]
  float* out = (float*)d_out;                // [2048,64]
  float* Ht  = (float*)d_ws;                 // [64,2048] = 512 KB scratch

  h_gemm_kernel<<<(FOUT * GN) / 256, 256, 0, stream>>>(X, W, b, Ht);

  dim3 grid(GN / JT2, FOUT / 8);             // 64 j-slabs x 8 feature-groups
  gconv_kernel<<<grid, 256, LDS_BYTES, stream>>>(A, mh, mf, Ht, out);
}